// QM9_NetGIN_26749056319924
// MI455X (gfx1250) — compile-verified
//
#include <hip/hip_runtime.h>

typedef __attribute__((ext_vector_type(2))) float v2f;
typedef __attribute__((ext_vector_type(8))) float v8f;

#define EMB 128
#define CC 12
#define BN_EPS 1e-5f

// ---------- helpers ----------

// monotone order-preserving float<->uint encoding (for atomic max on floats)
__device__ __forceinline__ unsigned encf(float f) {
  unsigned u = __float_as_uint(f);
  return (u & 0x80000000u) ? ~u : (u | 0x80000000u);
}
__device__ __forceinline__ float decf(unsigned u) {
  u = (u & 0x80000000u) ? (u & 0x7fffffffu) : ~u;
  return __uint_as_float(u);
}
__device__ __forceinline__ void atomAddF(float* p, float v) {
  __hip_atomic_fetch_add(p, v, __ATOMIC_RELAXED, __HIP_MEMORY_SCOPE_AGENT);
}

__device__ __forceinline__ v8f wmma_f32(v2f a, v2f b, v8f c) {
  return __builtin_amdgcn_wmma_f32_16x16x4_f32(false, a, false, b, (short)0, c,
                                               false, false);
}

// Stage one 16-K weight chunk, pair-interleaved:
//   buf[kp*288 + col*2 + 0] = W[kb*16 + 2kp][col]
//   buf[kp*288 + col*2 + 1] = W[kb*16 + 2kp+1][col]
// so a B fragment (v2f) is ONE aligned ds_load_b64.
// 288-float kp stride => lanes 0-15 (kp) and 16-31 (kp+1) hit disjoint bank halves.
__device__ __forceinline__ void stageW(const float* __restrict__ W, int kb,
                                       float* __restrict__ buf, int tid, int nthr) {
  for (int i = tid; i < 256; i += nthr) {          // 8 kp x 32 col-groups
    int kp  = i >> 5;
    int col = (i & 31) * 4;
    const float* w0 = W + (size_t)(kb * 16 + kp * 2) * EMB + col;
    float4 t0 = *(const float4*)w0;
    float4 t1 = *(const float4*)(w0 + EMB);
    float2* p = (float2*)&buf[kp * 288 + col * 2];
    p[0] = make_float2(t0.x, t1.x);
    p[1] = make_float2(t0.y, t1.y);
    p[2] = make_float2(t0.z, t1.z);
    p[3] = make_float2(t0.w, t1.w);
  }
}

// ---------- fused GEMM(+bias+BN+ReLU), fp32 WMMA ----------
// Y[n,0:128] = relu( bn( X @ W + b ) ),  W row-major [K][128].
// Block = 64 threads = 2 waves; each wave owns 32 rows (two 16-row A tiles that
// share every B fragment) x 128 cols -> 16 v8f accumulators (128 VGPRs).

__global__ __launch_bounds__(64) void k_dense128(
    const float* __restrict__ X, const float* __restrict__ W,
    const float* __restrict__ lb, const float* __restrict__ bs,
    const float* __restrict__ bb, const float* __restrict__ bm,
    const float* __restrict__ bv, float* __restrict__ Y, int nrows)
{
  __shared__ float xs[64 * 132];     // 64-row X tile, padded stride
  __shared__ float wp[2][8 * 288];   // double-buffered pair-interleaved W chunk
  const int tid  = threadIdx.x;
  const int row0 = blockIdx.x * 64;

  // stage X tile: 64 x 128 = 2048 float4
  for (int i = tid; i < 2048; i += 64) {
    int r = i >> 5;
    int c = (i & 31) * 4;
    float4 t = make_float4(0.f, 0.f, 0.f, 0.f);
    if (row0 + r < nrows)
      t = *(const float4*)(X + (size_t)(row0 + r) * EMB + c);
    float* p = &xs[r * 132 + c];
    p[0] = t.x; p[1] = t.y; p[2] = t.z; p[3] = t.w;
  }
  stageW(W, 0, wp[0], tid, 64);
  __syncthreads();

  const int lane = tid & 31;
  const int wv   = tid >> 5;
  const int m0   = wv * 32;          // wave's row offset within block tile
  const int lr   = lane & 15;
  const int hi   = lane >> 4;        // 0: K+{0,1}/rows M..M+7 ; 1: K+{2,3}/rows M+8..M+15

  v8f zero = {0.f,0.f,0.f,0.f,0.f,0.f,0.f,0.f};
  v8f acc0[8], acc1[8];
#pragma unroll
  for (int nt = 0; nt < 8; ++nt) { acc0[nt] = zero; acc1[nt] = zero; }

  for (int kb = 0; kb < 8; ++kb) {               // K blocks of 16
    const float* wc = wp[kb & 1];
    if (kb < 7) stageW(W, kb + 1, wp[(kb + 1) & 1], tid, 64);  // prefetch next chunk
#pragma unroll
    for (int k4 = 0; k4 < 4; ++k4) {
      const int ak = k4 * 4 + hi * 2;            // K within chunk for this lane half
      const int kp = (ak >> 1);                  // pair row in wc
      v2f a0 = *(const v2f*)&xs[(m0 + lr) * 132 + kb * 16 + ak];
      v2f a1 = *(const v2f*)&xs[(m0 + 16 + lr) * 132 + kb * 16 + ak];
#pragma unroll
      for (int nt = 0; nt < 8; ++nt) {
        v2f b = *(const v2f*)&wc[kp * 288 + (nt * 16 + lr) * 2];
        acc0[nt] = wmma_f32(a0, b, acc0[nt]);
        acc1[nt] = wmma_f32(a1, b, acc1[nt]);
      }
    }
    __syncthreads();
  }

  // fused bias + BN(eval) + ReLU epilogue
#pragma unroll
  for (int nt = 0; nt < 8; ++nt) {
    const int col = nt * 16 + lr;
    const float A = bs[col] * rsqrtf(bv[col] + BN_EPS);
    const float B = (lb[col] - bm[col]) * A + bb[col];
#pragma unroll
    for (int i = 0; i < 8; ++i) {
      int r0 = row0 + m0 + hi * 8 + i;           // C/D layout: vgpr i -> M=i / M=i+8
      if (r0 < nrows)
        Y[(size_t)r0 * EMB + col] = fmaxf(acc0[nt][i] * A + B, 0.f);
      int r1 = r0 + 16;
      if (r1 < nrows)
        Y[(size_t)r1 * EMB + col] = fmaxf(acc1[nt][i] * A + B, 0.f);
    }
  }
}

// K=16 variant for the input layer: X[n,16] @ W[16,128]
__global__ __launch_bounds__(64) void k_dense16(
    const float* __restrict__ X, const float* __restrict__ W,
    const float* __restrict__ lb, const float* __restrict__ bs,
    const float* __restrict__ bb, const float* __restrict__ bm,
    const float* __restrict__ bv, float* __restrict__ Y, int nrows)
{
  __shared__ float xs[64 * 20];
  __shared__ float wp[8 * 288];
  const int tid  = threadIdx.x;
  const int row0 = blockIdx.x * 64;

  for (int i = tid; i < 256; i += 64) {          // 64 x 16 = 256 float4
    int r = i >> 2;
    int c = (i & 3) * 4;
    float4 t = make_float4(0.f, 0.f, 0.f, 0.f);
    if (row0 + r < nrows)
      t = *(const float4*)(X + (size_t)(row0 + r) * 16 + c);
    float* p = &xs[r * 20 + c];
    p[0] = t.x; p[1] = t.y; p[2] = t.z; p[3] = t.w;
  }
  stageW(W, 0, wp, tid, 64);
  __syncthreads();

  const int lane = tid & 31;
  const int wv   = tid >> 5;
  const int m0   = wv * 32;
  const int lr   = lane & 15;
  const int hi   = lane >> 4;

  v8f zero = {0.f,0.f,0.f,0.f,0.f,0.f,0.f,0.f};
  v8f acc0[8], acc1[8];
#pragma unroll
  for (int nt = 0; nt < 8; ++nt) { acc0[nt] = zero; acc1[nt] = zero; }

#pragma unroll
  for (int k4 = 0; k4 < 4; ++k4) {
    const int ak = k4 * 4 + hi * 2;
    const int kp = (ak >> 1);
    v2f a0 = *(const v2f*)&xs[(m0 + lr) * 20 + ak];
    v2f a1 = *(const v2f*)&xs[(m0 + 16 + lr) * 20 + ak];
#pragma unroll
    for (int nt = 0; nt < 8; ++nt) {
      v2f b = *(const v2f*)&wp[kp * 288 + (nt * 16 + lr) * 2];
      acc0[nt] = wmma_f32(a0, b, acc0[nt]);
      acc1[nt] = wmma_f32(a1, b, acc1[nt]);
    }
  }

#pragma unroll
  for (int nt = 0; nt < 8; ++nt) {
    const int col = nt * 16 + lr;
    const float A = bs[col] * rsqrtf(bv[col] + BN_EPS);
    const float B = (lb[col] - bm[col]) * A + bb[col];
#pragma unroll
    for (int i = 0; i < 8; ++i) {
      int r0 = row0 + m0 + hi * 8 + i;
      if (r0 < nrows)
        Y[(size_t)r0 * EMB + col] = fmaxf(acc0[nt][i] * A + B, 0.f);
      int r1 = r0 + 16;
      if (r1 < nrows)
        Y[(size_t)r1 * EMB + col] = fmaxf(acc1[nt][i] * A + B, 0.f);
    }
  }
}

// ---------- elementwise / graph kernels ----------

// y = (1 + eps[l]) * x   (float4 granularity)
__global__ void k_scale(const float* __restrict__ x, float* __restrict__ y,
                        const float* __restrict__ eps, int l, int n4)
{
  int i = blockIdx.x * blockDim.x + threadIdx.x;
  if (i >= n4) return;
  float s = 1.f + eps[l];
  float4 v = ((const float4*)x)[i];
  v.x *= s; v.y *= s; v.z *= s; v.w *= s;
  ((float4*)y)[i] = v;
}

// one wave per edge: out[dst] += h[src] (128 features, float4 per lane)
__global__ void k_scatter(const float* __restrict__ h, const int* __restrict__ ei,
                          float* __restrict__ out, int E)
{
  int gid = blockIdx.x * blockDim.x + threadIdx.x;
  int e = gid >> 5;
  int lane = gid & 31;
  if (e >= E) return;
  int s = ei[e];
  int d = ei[E + e];
  float4 v = *(const float4*)(h + (size_t)s * EMB + lane * 4);
  float* o = out + (size_t)d * EMB + lane * 4;
  atomAddF(o + 0, v.x);
  atomAddF(o + 1, v.y);
  atomAddF(o + 2, v.z);
  atomAddF(o + 3, v.w);
}

__global__ void k_fill(unsigned* __restrict__ p, int n, unsigned val)
{
  int i = blockIdx.x * blockDim.x + threadIdx.x;
  if (i < n) p[i] = val;
}

// pooled_enc[batch[n]*128+f] = max(enc(h[n,f]))
__global__ void k_poolmax(const float* __restrict__ h, const int* __restrict__ batch,
                          unsigned* __restrict__ pooled, int N)
{
  int idx = blockIdx.x * blockDim.x + threadIdx.x;
  if (idx >= N * EMB) return;
  int n = idx >> 7;
  int f = idx & 127;
  atomicMax(&pooled[(size_t)batch[n] * EMB + f], encf(h[idx]));
}

// initial readout: z[n,c] = h[n,:]@W[:,c] + b[c]; atomic max into outEnc per graph
__global__ void k_initread(const float* __restrict__ h, const int* __restrict__ batch,
                           const float* __restrict__ W, const float* __restrict__ b,
                           unsigned* __restrict__ outEnc, int N)
{
  int idx = blockIdx.x * blockDim.x + threadIdx.x;
  int n = idx >> 4;
  int c = idx & 15;
  if (n >= N || c >= CC) return;
  float s = b[c];
  const float* hr = h + (size_t)n * EMB;
  for (int k = 0; k < EMB; ++k) s += hr[k] * W[k * CC + c];
  atomicMax(&outEnc[(size_t)batch[n] * CC + c], encf(s));
}

__global__ void k_decode(const unsigned* __restrict__ in, float* __restrict__ out, int n)
{
  int i = blockIdx.x * blockDim.x + threadIdx.x;
  if (i < n) out[i] = decf(in[i]);
}

// out[g,c] += dec(pooled[g,:]) @ W[:,c] + b[c]
__global__ void k_proj(const unsigned* __restrict__ pooled, const float* __restrict__ W,
                       const float* __restrict__ b, float* __restrict__ out, int G)
{
  int idx = blockIdx.x * blockDim.x + threadIdx.x;
  int g = idx >> 4;
  int c = idx & 15;
  if (g >= G || c >= CC) return;
  float s = b[c];
  const unsigned* pr = pooled + (size_t)g * EMB;
  for (int k = 0; k < EMB; ++k) s += decf(pr[k]) * W[k * CC + c];
  out[(size_t)g * CC + c] += s;
}

// ---------- launch ----------

extern "C" void kernel_launch(void* const* d_in, const int* in_sizes, int n_in,
                              void* d_out, int out_size, void* d_ws, size_t ws_size,
                              hipStream_t stream)
{
  const float* x     = (const float*)d_in[0];
  const int*   ei    = (const int*)  d_in[1];
  const int*   batch = (const int*)  d_in[2];
  const float* iw1   = (const float*)d_in[3];
  const float* ib1   = (const float*)d_in[4];
  const float* ibn1s = (const float*)d_in[5];
  const float* ibn1b = (const float*)d_in[6];
  const float* ibn1m = (const float*)d_in[7];
  const float* ibn1v = (const float*)d_in[8];
  const float* iw2   = (const float*)d_in[9];
  const float* ib2   = (const float*)d_in[10];
  const float* ibn2s = (const float*)d_in[11];
  const float* ibn2b = (const float*)d_in[12];
  const float* ibn2m = (const float*)d_in[13];
  const float* ibn2v = (const float*)d_in[14];
  const float* ilw   = (const float*)d_in[15];
  const float* ilb   = (const float*)d_in[16];
  const float* eps   = (const float*)d_in[17];
  const float* lw1   = (const float*)d_in[18];
  const float* lb1   = (const float*)d_in[19];
  const float* lbn1s = (const float*)d_in[20];
  const float* lbn1b = (const float*)d_in[21];
  const float* lbn1m = (const float*)d_in[22];
  const float* lbn1v = (const float*)d_in[23];
  const float* lw2   = (const float*)d_in[24];
  const float* lb2   = (const float*)d_in[25];
  const float* lbn2s = (const float*)d_in[26];
  const float* lbn2b = (const float*)d_in[27];
  const float* lbn2m = (const float*)d_in[28];
  const float* lbn2v = (const float*)d_in[29];
  const float* llw   = (const float*)d_in[30];
  const float* llb   = (const float*)d_in[31];

  const int N = in_sizes[0] / 16;   // F_IN = 16
  const int E = in_sizes[1] / 2;
  const int G = out_size / CC;
  const int L = in_sizes[17];

  float*    bufH   = (float*)d_ws;
  float*    bufT   = bufH + (size_t)N * EMB;
  unsigned* pooled = (unsigned*)(bufT + (size_t)N * EMB);
  unsigned* outEnc = pooled + (size_t)G * EMB;
  float*    out    = (float*)d_out;

  const int dblocks = (N + 63) / 64;

  // h1 = relu(bn1(x @ iw1 + ib1))  -> bufT
  k_dense16<<<dblocks, 64, 0, stream>>>(x, iw1, ib1, ibn1s, ibn1b, ibn1m, ibn1v, bufT, N);
  // h  = relu(bn2(h1 @ iw2 + ib2)) -> bufH
  k_dense128<<<dblocks, 64, 0, stream>>>(bufT, iw2, ib2, ibn2s, ibn2b, ibn2m, ibn2v, bufH, N);

  // initial readout: out = segment_max(h @ ilw + ilb)
  k_fill<<<(G * CC + 255) / 256, 256, 0, stream>>>(outEnc, G * CC, 0u);
  k_initread<<<(N * 16 + 255) / 256, 256, 0, stream>>>(bufH, batch, ilw, ilb, outEnc, N);
  k_decode<<<(G * CC + 255) / 256, 256, 0, stream>>>(outEnc, out, G * CC);

  for (int l = 0; l < L; ++l) {
    const int n4 = N * EMB / 4;
    // agg = (1+eps)*h + scatter_add(h[src] -> dst)
    k_scale<<<(n4 + 255) / 256, 256, 0, stream>>>(bufH, bufT, eps, l, n4);
    k_scatter<<<(E * 32 + 255) / 256, 256, 0, stream>>>(bufH, ei, bufT, E);
    // two fused GEMM+BN+ReLU layers
    k_dense128<<<dblocks, 64, 0, stream>>>(bufT, lw1 + (size_t)l * EMB * EMB, lb1 + l * EMB,
        lbn1s + l * EMB, lbn1b + l * EMB, lbn1m + l * EMB, lbn1v + l * EMB, bufH, N);
    k_dense128<<<dblocks, 64, 0, stream>>>(bufH, lw2 + (size_t)l * EMB * EMB, lb2 + l * EMB,
        lbn2s + l * EMB, lbn2b + l * EMB, lbn2m + l * EMB, lbn2v + l * EMB, bufT, N);
    { float* t = bufH; bufH = bufT; bufT = t; }   // new h lives in bufH

    // per-layer readout: out += segment_max(h) @ llw[l] + llb[l]
    k_fill<<<(G * EMB + 255) / 256, 256, 0, stream>>>(pooled, G * EMB, 0u);
    k_poolmax<<<(N * EMB + 255) / 256, 256, 0, stream>>>(bufH, batch, pooled, N);
    k_proj<<<(G * 16 + 255) / 256, 256, 0, stream>>>(pooled, llw + (size_t)l * EMB * CC,
                                                     llb + l * CC, out, G);
  }
}